// ImprovedWordGCN_67817533604035
// MI455X (gfx1250) — compile-verified
//
#include <hip/hip_runtime.h>
#include <hip/hip_bf16.h>

// ---------------- problem constants (match reference) ----------------
#define NUM_WORDS   30000
#define NUM_DOCS    15000
#define HIDDEN      300
#define HALF_HIDDEN 150
#define RES_ALPHA   0.7f
#define LN_EPS      1e-5f

typedef __attribute__((ext_vector_type(2))) float v2f;
typedef __attribute__((ext_vector_type(8))) float v8f;

// ---------------------------------------------------------------------
// zero-fill (grid-stride)
// ---------------------------------------------------------------------
__global__ void zero_f32(float* __restrict__ p, size_t n) {
  size_t i = (size_t)blockIdx.x * blockDim.x + threadIdx.x;
  size_t stride = (size_t)gridDim.x * blockDim.x;
  for (; i < n; i += stride) p[i] = 0.0f;
}

// ---------------------------------------------------------------------
// SpMM via COO + atomics: out[rows[e], :] += vals[e] * H[cols[e], :]
// One wave32 per nonzero; lanes stride the HIDDEN dim (coalesced).
// ---------------------------------------------------------------------
__global__ __launch_bounds__(256)
void spmm_atomic(const int* __restrict__ rows, const int* __restrict__ cols,
                 const float* __restrict__ vals, const float* __restrict__ H,
                 float* __restrict__ out, int nnz) {
  int gwid = (int)((blockIdx.x * blockDim.x + threadIdx.x) >> 5);
  int lane = threadIdx.x & 31;
  if (gwid >= nnz) return;
  int r = rows[gwid];
  int c = cols[gwid];
  float v = vals[gwid];
  const float* __restrict__ src = H + (size_t)c * HIDDEN;
  float*       __restrict__ dst = out + (size_t)r * HIDDEN;
  for (int j = lane; j < HIDDEN; j += 32)
    atomicAdd(dst + j, v * src[j]);
}

// ---------------------------------------------------------------------
// Dense GEMM on the fp32 WMMA path: C = act(A[MxK] @ W[KxN] + bias)
// W is row-major [K, N]. Each wave computes a 64x16 tile (4 stacked
// 16x16 accumulators), reusing the B fragment across the 4 M-subtiles.
//
// Hot-loop strategy: NO per-element predication. K must be a multiple
// of 4 (it is 300 at every call site). M/N edge tiles CLAMP their load
// indices to the last valid row/col: clamped loads touch valid memory,
// and the garbage they produce only affects out-of-range output
// rows/cols, which the store epilogue discards. A fragments are loaded
// as one 8-byte global_load_b64 per subtile (kk is always even so the
// v2f load is 8-byte aligned).
//
// V_WMMA_F32_16X16X4_F32 operand layouts (ISA 7.12.2):
//   A 16x4 : lane<16 -> M=lane,   v2f = {A[m][k0+0], A[m][k0+1]}
//            lane>=16-> M=lane-16, v2f = {A[m][k0+2], A[m][k0+3]}
//   B 4x16 : lane<16 -> N=lane,   v2f = {W[k0+0][n], W[k0+1][n]}
//            lane>=16-> N=lane-16, v2f = {W[k0+2][n], W[k0+3][n]}
//   C 16x16: VGPR j -> row = 8*(lane>=16)+j, col = lane&15
// ---------------------------------------------------------------------
__global__ __launch_bounds__(256)
void gemm_wmma_f32(const float* __restrict__ A, const float* __restrict__ W,
                   const float* __restrict__ bias, float* __restrict__ C,
                   int M, int K, int N, int relu) {
  const int lane = threadIdx.x & 31;
  const int widb = threadIdx.x >> 5;
  const int tiles_m = (M + 63) >> 6;
  const int tiles_n = (N + 15) >> 4;
  const int wave = (int)blockIdx.x * (blockDim.x >> 5) + widb;
  if (wave >= tiles_m * tiles_n) return;

  const int tn = wave % tiles_n;
  const int tm = wave / tiles_n;
  const int m0 = tm << 6;
  const int n0 = tn << 4;

  const int col    = n0 + (lane & 15);        // output/B column for this lane
  const bool colok = (col < N);
  const int colc   = colok ? col : (N - 1);   // clamped for loads
  const int khalf  = (lane >> 4) << 1;        // 0 (lanes 0-15) or 2 (lanes 16-31)
  const int mrow   = lane & 15;

  // clamped A-row indices for the 4 stacked M-subtiles
  const int mr0 = min(m0 +  0 + mrow, M - 1);
  const int mr1 = min(m0 + 16 + mrow, M - 1);
  const int mr2 = min(m0 + 32 + mrow, M - 1);
  const int mr3 = min(m0 + 48 + mrow, M - 1);

  const float* __restrict__ ap0 = A + (size_t)mr0 * K + khalf;
  const float* __restrict__ ap1 = A + (size_t)mr1 * K + khalf;
  const float* __restrict__ ap2 = A + (size_t)mr2 * K + khalf;
  const float* __restrict__ ap3 = A + (size_t)mr3 * K + khalf;
  const float* __restrict__ bp  = W + (size_t)khalf * N + colc;
  const size_t bstep = (size_t)4 * N;

  v8f acc0 = {}, acc1 = {}, acc2 = {}, acc3 = {};

#pragma unroll 2
  for (int k0 = 0; k0 < K; k0 += 4) {
    v2f b;
    b.x = bp[0];
    b.y = bp[N];
    const v2f a0 = *(const v2f*)ap0;
    const v2f a1 = *(const v2f*)ap1;
    const v2f a2 = *(const v2f*)ap2;
    const v2f a3 = *(const v2f*)ap3;
    acc0 = __builtin_amdgcn_wmma_f32_16x16x4_f32(false, a0, false, b, (short)0, acc0, false, false);
    acc1 = __builtin_amdgcn_wmma_f32_16x16x4_f32(false, a1, false, b, (short)0, acc1, false, false);
    acc2 = __builtin_amdgcn_wmma_f32_16x16x4_f32(false, a2, false, b, (short)0, acc2, false, false);
    acc3 = __builtin_amdgcn_wmma_f32_16x16x4_f32(false, a3, false, b, (short)0, acc3, false, false);
    ap0 += 4; ap1 += 4; ap2 += 4; ap3 += 4;
    bp  += bstep;
  }

  // -------- epilogue: bias + relu + guarded stores --------
  const float bval = (bias != nullptr && colok) ? bias[col] : 0.0f;
  const int rtop = (lane >> 4) << 3;  // +8 for upper half-lanes

  auto store_tile = [&](const v8f& acc, int mbase) {
#pragma unroll
    for (int j = 0; j < 8; ++j) {
      const int row = mbase + rtop + j;
      if (row < M && colok) {
        float v = acc[j] + bval;
        if (relu) v = fmaxf(v, 0.0f);
        C[(size_t)row * N + col] = v;
      }
    }
  };
  store_tile(acc0, m0 +  0);
  store_tile(acc1, m0 + 16);
  store_tile(acc2, m0 + 32);
  store_tile(acc3, m0 + 48);
}

// ---------------------------------------------------------------------
// Fused residual + LayerNorm + (word_H + emb) producer.
// y   = (1-alpha)*emb + alpha*H
// wH  = (y - mu)*rsqrt(var+eps)*g + b
// out = wH + emb            (feeds the single fused doc SpMM)
// One wave32 per word row; shuffle reductions.
// ---------------------------------------------------------------------
__global__ __launch_bounds__(256)
void resid_ln_wsum(const float* __restrict__ emb, const float* __restrict__ H,
                   const float* __restrict__ g, const float* __restrict__ b,
                   float* __restrict__ wsum) {
  int wave = (int)((blockIdx.x * blockDim.x + threadIdx.x) >> 5);
  int lane = threadIdx.x & 31;
  if (wave >= NUM_WORDS) return;

  const float* __restrict__ e = emb + (size_t)wave * HIDDEN;
  const float* __restrict__ h = H   + (size_t)wave * HIDDEN;
  float*       __restrict__ o = wsum + (size_t)wave * HIDDEN;

  float x[10];  // ceil(300/32) = 10 elements max per lane
  int cnt = 0;
  float s = 0.0f;
  for (int j = lane; j < HIDDEN; j += 32) {
    float v = (1.0f - RES_ALPHA) * e[j] + RES_ALPHA * h[j];
    x[cnt++] = v;
    s += v;
  }
#pragma unroll
  for (int off = 16; off > 0; off >>= 1) s += __shfl_xor(s, off, 32);
  const float mu = s * (1.0f / HIDDEN);

  float vs = 0.0f;
  for (int t = 0; t < cnt; ++t) { float d = x[t] - mu; vs += d * d; }
#pragma unroll
  for (int off = 16; off > 0; off >>= 1) vs += __shfl_xor(vs, off, 32);
  const float rstd = rsqrtf(vs * (1.0f / HIDDEN) + LN_EPS);

  cnt = 0;
  for (int j = lane; j < HIDDEN; j += 32) {
    float w = (x[cnt++] - mu) * rstd * g[j] + b[j];
    o[j] = w + e[j];
  }
}

// ---------------------------------------------------------------------
// Final classifier head: logits[d, c] = h2[d, :150] @ clf_W[:, c] + clf_b[c]
// N=2 — scalar kernel, one thread per doc. clf_W lives in L2/registers.
// ---------------------------------------------------------------------
__global__ __launch_bounds__(256)
void clf_head(const float* __restrict__ h2, const float* __restrict__ cw,
              const float* __restrict__ cb, float* __restrict__ out, int ndocs) {
  int d = (int)(blockIdx.x * blockDim.x + threadIdx.x);
  if (d >= ndocs) return;
  const float* __restrict__ h = h2 + (size_t)d * HALF_HIDDEN;
  float a0 = cb[0], a1 = cb[1];
  for (int k = 0; k < HALF_HIDDEN; ++k) {
    float v = h[k];
    a0 = fmaf(v, cw[2 * k + 0], a0);
    a1 = fmaf(v, cw[2 * k + 1], a1);
  }
  out[2 * d + 0] = a0;
  out[2 * d + 1] = a1;
}

// ---------------------------------------------------------------------
// Host-side orchestration (graph-capture safe: kernels only, ws scratch)
// ---------------------------------------------------------------------
extern "C" void kernel_launch(void* const* d_in, const int* in_sizes, int n_in,
                              void* d_out, int out_size, void* d_ws, size_t ws_size,
                              hipStream_t stream) {
  const int*   A_rows = (const int*)  d_in[0];
  const int*   A_cols = (const int*)  d_in[1];
  const float* A_vals = (const float*)d_in[2];
  const int*   X_rows = (const int*)  d_in[3];
  const int*   X_cols = (const int*)  d_in[4];
  const float* X_vals = (const float*)d_in[5];
  const float* emb    = (const float*)d_in[6];
  const float* W1     = (const float*)d_in[7];
  const float* W2     = (const float*)d_in[8];
  const float* W3     = (const float*)d_in[9];
  const float* ln_g   = (const float*)d_in[10];
  const float* ln_b   = (const float*)d_in[11];
  const float* mlp_W1 = (const float*)d_in[12];
  const float* mlp_b1 = (const float*)d_in[13];
  const float* mlp_W2 = (const float*)d_in[14];
  const float* mlp_b2 = (const float*)d_in[15];
  const float* clf_W  = (const float*)d_in[16];
  const float* clf_b  = (const float*)d_in[17];
  float* out = (float*)d_out;

  const int NNZ_A = in_sizes[0];
  const int NNZ_X = in_sizes[3];

  // workspace carve-up (floats): 3x word-sized + doc buffers = ~153 MB
  const size_t WORD_ELEMS = (size_t)NUM_WORDS * HIDDEN;       // 9,000,000
  const size_t DOC_ELEMS  = (size_t)NUM_DOCS  * HIDDEN;       // 4,500,000
  float* buf0 = (float*)d_ws;            // SpMM accum / H3
  float* buf1 = buf0 + WORD_ELEMS;       // H1 / S3 / wsum
  float* buf2 = buf1 + WORD_ELEMS;       // H2
  float* buf3 = buf2 + WORD_ELEMS;       // doc_H
  float* buf4 = buf3 + DOC_ELEMS;        // h1
  float* buf5 = buf4 + DOC_ELEMS;        // h2

  const dim3 B(256);
  const unsigned ZG = 2048;  // grid-stride zero fill
  auto spmm_blocks = [](int nnz) { return (unsigned)((nnz + 7) / 8); };  // 8 waves/block
  auto gemm_blocks = [](int M, int N) {
    int tiles = ((M + 63) >> 6) * ((N + 15) >> 4);
    return (unsigned)((tiles + 7) / 8);                                   // 8 waves/block
  };

  // ---- GCN layer 1: H1 = relu((A @ emb) @ W1) ----
  zero_f32<<<ZG, B, 0, stream>>>(buf0, WORD_ELEMS);
  spmm_atomic<<<spmm_blocks(NNZ_A), B, 0, stream>>>(A_rows, A_cols, A_vals, emb, buf0, NNZ_A);
  gemm_wmma_f32<<<gemm_blocks(NUM_WORDS, HIDDEN), B, 0, stream>>>(
      buf0, W1, nullptr, buf1, NUM_WORDS, HIDDEN, HIDDEN, 1);

  // ---- GCN layer 2: H2 = relu((A @ H1) @ W2) ----
  zero_f32<<<ZG, B, 0, stream>>>(buf0, WORD_ELEMS);
  spmm_atomic<<<spmm_blocks(NNZ_A), B, 0, stream>>>(A_rows, A_cols, A_vals, buf1, buf0, NNZ_A);
  gemm_wmma_f32<<<gemm_blocks(NUM_WORDS, HIDDEN), B, 0, stream>>>(
      buf0, W2, nullptr, buf2, NUM_WORDS, HIDDEN, HIDDEN, 1);

  // ---- GCN layer 3: H3 = relu((A @ H2) @ W3) ----
  zero_f32<<<ZG, B, 0, stream>>>(buf1, WORD_ELEMS);
  spmm_atomic<<<spmm_blocks(NNZ_A), B, 0, stream>>>(A_rows, A_cols, A_vals, buf2, buf1, NNZ_A);
  gemm_wmma_f32<<<gemm_blocks(NUM_WORDS, HIDDEN), B, 0, stream>>>(
      buf1, W3, nullptr, buf0, NUM_WORDS, HIDDEN, HIDDEN, 1);

  // ---- residual + LayerNorm, fused with (word_H + emb) ----
  resid_ln_wsum<<<(NUM_WORDS + 7) / 8, B, 0, stream>>>(emb, buf0, ln_g, ln_b, buf1);

  // ---- doc aggregation: doc_H = X @ (word_H + emb)  (fused: one SpMM) ----
  zero_f32<<<ZG, B, 0, stream>>>(buf3, DOC_ELEMS);
  spmm_atomic<<<spmm_blocks(NNZ_X), B, 0, stream>>>(X_rows, X_cols, X_vals, buf1, buf3, NNZ_X);

  // ---- MLP classifier ----
  gemm_wmma_f32<<<gemm_blocks(NUM_DOCS, HIDDEN), B, 0, stream>>>(
      buf3, mlp_W1, mlp_b1, buf4, NUM_DOCS, HIDDEN, HIDDEN, 1);
  gemm_wmma_f32<<<gemm_blocks(NUM_DOCS, HALF_HIDDEN), B, 0, stream>>>(
      buf4, mlp_W2, mlp_b2, buf5, NUM_DOCS, HIDDEN, HALF_HIDDEN, 1);
  clf_head<<<(NUM_DOCS + 255) / 256, B, 0, stream>>>(buf5, clf_W, clf_b, out, NUM_DOCS);

  (void)n_in; (void)out_size; (void)ws_size;
}